// STDF_12713103196504
// MI455X (gfx1250) — compile-verified
//
#include <hip/hip_runtime.h>
#include <math.h>

typedef __attribute__((ext_vector_type(2))) float v2f;
typedef __attribute__((ext_vector_type(8))) float v8f;

#define TPB 256
#define RSB 136   // B-panel row stride in floats (128 + 8 pad: 2-row offset = 16 banks)

// ---------------------------------------------------------------------------
// Implicit-GEMM 3x3 conv (pad=1) via V_WMMA_F32_16X16X4_F32 (exact fp32).
//   Block: M = 16 output channels, N = 128 flat output pixels (8 waves x 16).
//   K = Cin*9, consumed in chunks of 4 input channels = 36 k-rows = 9 WMMA
//   steps; B-panel (im2col) staged to LDS per chunk so the WMMA loop is pure
//   ds_load + v_wmma. All H/W are powers of two -> flat->(oy,ox) is shift/mask.
// ---------------------------------------------------------------------------
template<int STRIDE, bool RELU>
__global__ __launch_bounds__(TPB)
void conv3x3_wmma(const float* __restrict__ x, const float* __restrict__ w,
                  const float* __restrict__ bias, float* __restrict__ y,
                  int Cin, int H, int W, int Cout, int HoWo, int woShift)
{
    extern __shared__ float lds[];
    const int K    = Cin * 9;
    const int nchk = (Cin + 3) >> 2;
    const int Kp   = nchk * 36;            // zero-padded K rows
    float* ldsw = lds;                     // [Kp][16]   weights, [k][m]
    float* ldsb = lds + Kp * 16;           // [36][RSB]  im2col chunk

    const int co0 = blockIdx.y * 16;
    const int b   = blockIdx.z;
    const int Wo  = 1 << woShift;

    // Stage this block's 16 output-channel weight rows (zero-padded).
    for (int i = threadIdx.x; i < Kp * 16; i += TPB) {
        int k = i >> 4, m = i & 15;
        float v = 0.f;
        if (k < K && (co0 + m) < Cout)
            v = w[(size_t)(co0 + m) * K + k];   // w: (Cout, Cin, 3, 3)
        ldsw[i] = v;
    }

    const int lane = threadIdx.x & 31;
    const int wv   = threadIdx.x >> 5;
    const int half = lane >> 4;            // selects K rows k0,k0+1 vs k0+2,k0+3
    const int idx  = lane & 15;            // M (A) / N (B,D) index
    const int nn   = wv * 16 + idx;        // this lane's pixel column in block
    const int base = blockIdx.x * 128;     // flat output pixel base

    const float* xb = x + (size_t)b * Cin * H * W;

    v8f acc = {0.f, 0.f, 0.f, 0.f, 0.f, 0.f, 0.f, 0.f};

    for (int c0 = 0; c0 < Cin; c0 += 4) {
        __syncthreads();                   // ldsb reuse / first-iter ldsw fence
        // Stage im2col chunk: 36 k-rows x 128 pixels (zero-padded, coalesced in n).
        for (int i = threadIdx.x; i < 36 * 128; i += TPB) {
            int kr = i >> 7, n = i & 127;
            int ci = kr / 9, rs = kr - 9 * ci;
            int r  = rs / 3, s  = rs - 3 * r;
            int c  = c0 + ci;
            int flat = base + n;
            int oy = flat >> woShift, ox = flat & (Wo - 1);
            int iy = oy * STRIDE + r - 1;
            int ix = ox * STRIDE + s - 1;
            float v = 0.f;
            if (c < Cin && iy >= 0 && iy < H && ix >= 0 && ix < W)
                v = xb[((size_t)c * H + iy) * W + ix];
            ldsb[kr * RSB + n] = v;
        }
        __syncthreads();

        const int kg0 = c0 * 9;
        #pragma unroll
        for (int st = 0; st < 9; ++st) {
            int kr = st * 4 + 2 * half;
            v2f a, bb;
            a[0]  = ldsw[(kg0 + kr)     * 16 + idx];   // A[m=idx][k]
            a[1]  = ldsw[(kg0 + kr + 1) * 16 + idx];
            bb[0] = ldsb[(kr)     * RSB + nn];         // B[k][n=nn]
            bb[1] = ldsb[(kr + 1) * RSB + nn];
            acc = __builtin_amdgcn_wmma_f32_16x16x4_f32(false, a, false, bb,
                                                        (short)0, acc, false, false);
        }
    }

    // D layout: vgpr i -> M = i + 8*half, N = idx (pixel = base + nn)
    #pragma unroll
    for (int i = 0; i < 8; ++i) {
        int co = co0 + i + 8 * half;
        if (co < Cout) {
            float v = acc[i] + bias[co];
            if (RELU) v = fmaxf(v, 0.f);
            y[((size_t)b * Cout + co) * HoWo + base + nn] = v;
        }
    }
}

// ---------------------------------------------------------------------------
// Transposed conv 4x4 stride 2 "pad 1" (output = 2H x 2W), gather form:
//  y[o,oy,ox] = b[o] + sum_{i,r,s: oy+r-2 even, ox+s-2 even, in range}
//               x[i,(oy+r-2)/2,(ox+s-2)/2] * w[i,o,3-r,3-s];  w: (Cin,Cout,4,4)
// ---------------------------------------------------------------------------
__global__ __launch_bounds__(TPB)
void convT4x4(const float* __restrict__ x, const float* __restrict__ w,
              const float* __restrict__ bias, float* __restrict__ y,
              int B, int Cin, int H, int W, int Cout)
{
    const int H2 = 2 * H, W2 = 2 * W;
    const size_t total = (size_t)B * Cout * H2 * W2;
    size_t t = (size_t)blockIdx.x * TPB + threadIdx.x;
    if (t >= total) return;
    int ox = (int)(t % W2);
    int oy = (int)((t / W2) % H2);
    int o  = (int)((t / ((size_t)W2 * H2)) % Cout);
    int b  = (int)(t / ((size_t)W2 * H2 * Cout));

    float acc = bias[o];
    const float* xb = x + (size_t)b * Cin * H * W;
    for (int r = 0; r < 4; ++r) {
        int p = oy + r - 2;
        if (p & 1) continue;
        int iy = p >> 1;
        if (iy < 0 || iy >= H) continue;
        for (int s = 0; s < 4; ++s) {
            int q = ox + s - 2;
            if (q & 1) continue;
            int ix = q >> 1;
            if (ix < 0 || ix >= W) continue;
            const float* xp = xb + (size_t)iy * W + ix;
            const float* wp = w + ((size_t)o * 4 + (3 - r)) * 4 + (3 - s);
            for (int c = 0; c < Cin; ++c)
                acc += xp[(size_t)c * H * W] * wp[(size_t)c * Cout * 16];
        }
    }
    y[t] = acc;
}

// ---------------------------------------------------------------------------
// Channel concat: out[b] = [a (C1 ch) ; bsrc (C2 ch)]
// ---------------------------------------------------------------------------
__global__ __launch_bounds__(TPB)
void concat2(const float* __restrict__ a, const float* __restrict__ bsrc,
             float* __restrict__ o, int C1, int C2, int HW, int B)
{
    size_t total = (size_t)B * (C1 + C2) * HW;
    size_t t = (size_t)blockIdx.x * TPB + threadIdx.x;
    if (t >= total) return;
    int hw = (int)(t % HW);
    int c  = (int)((t / HW) % (C1 + C2));
    int b  = (int)(t / ((size_t)HW * (C1 + C2)));
    o[t] = (c < C1) ? a[((size_t)b * C1 + c) * HW + hw]
                    : bsrc[((size_t)b * C2 + (c - C1)) * HW + hw];
}

// ---------------------------------------------------------------------------
// Deformable sampling + dc einsum + bias, fused (sigmoid applied to mask).
// One thread per output pixel; 64 fp32 accumulators; dc_w/dc_b in LDS.
// ---------------------------------------------------------------------------
#define OUTC 64
#define INC  7
#define KK   9

__device__ __forceinline__ float samp1(const float* __restrict__ xc,
                                       int iy, int ix, int H, int W)
{
    bool valid = (iy >= 0) & (iy < H) & (ix >= 0) & (ix < W);
    int yc  = min(max(iy, 0), H - 1);
    int xc2 = min(max(ix, 0), W - 1);
    float v = xc[(size_t)yc * W + xc2];
    return valid ? v : 0.f;
}

__global__ __launch_bounds__(TPB)
void deform_out(const float* __restrict__ xf, const float* __restrict__ om,
                const float* __restrict__ dcw, const float* __restrict__ dcb,
                float* __restrict__ y, int B, int H, int W)
{
    __shared__ float sdw[OUTC * INC * KK];
    __shared__ float sdb[OUTC];
    for (int i = threadIdx.x; i < OUTC * INC * KK; i += TPB) sdw[i] = dcw[i];
    if (threadIdx.x < OUTC) sdb[threadIdx.x] = dcb[threadIdx.x];
    __syncthreads();

    const size_t HW = (size_t)H * W;
    size_t t = (size_t)blockIdx.x * TPB + threadIdx.x;
    if (t >= (size_t)B * HW) return;
    int x0 = (int)(t % W);
    int y0 = (int)((t / W) % H);
    int b  = (int)(t / HW);

    float acc[OUTC];
    #pragma unroll
    for (int o = 0; o < OUTC; ++o) acc[o] = 0.f;

    const float* xb  = xf + (size_t)b * INC * HW;
    const float* ob  = om + (size_t)b * (INC * 3 * KK) * HW;
    const size_t pix = (size_t)y0 * W + x0;

    for (int c = 0; c < INC; ++c) {
        const float* xc = xb + (size_t)c * HW;
        for (int k = 0; k < KK; ++k) {
            float dy = ob[(size_t)((c * KK + k) * 2 + 0) * HW + pix];
            float dx = ob[(size_t)((c * KK + k) * 2 + 1) * HW + pix];
            float m  = ob[(size_t)(INC * 2 * KK + c * KK + k) * HW + pix];
            m = 1.f / (1.f + expf(-m));
            float ys = dy + (float)y0 + (float)(k / 3 - 1);
            float xs = dx + (float)x0 + (float)(k % 3 - 1);
            float yf = floorf(ys), xff = floorf(xs);
            float wy = ys - yf,    wx  = xs - xff;
            int iy = (int)yf, ix = (int)xff;
            float v00 = samp1(xc, iy,     ix,     H, W);
            float v01 = samp1(xc, iy,     ix + 1, H, W);
            float v10 = samp1(xc, iy + 1, ix,     H, W);
            float v11 = samp1(xc, iy + 1, ix + 1, H, W);
            float s = (v00 * (1.f - wy) * (1.f - wx) + v01 * (1.f - wy) * wx +
                       v10 * wy * (1.f - wx)         + v11 * wy * wx) * m;
            const float* wrow = sdw + (c * KK + k);
            #pragma unroll
            for (int o = 0; o < OUTC; ++o)
                acc[o] += s * wrow[o * (INC * KK)];
        }
    }
    float* yb = y + (size_t)b * OUTC * HW;
    #pragma unroll
    for (int o = 0; o < OUTC; ++o)
        yb[(size_t)o * HW + pix] = acc[o] + sdb[o];
}

// ---------------------------------------------------------------------------
// Host orchestration
// ---------------------------------------------------------------------------
extern "C" void kernel_launch(void* const* d_in, const int* in_sizes, int n_in,
                              void* d_out, int out_size, void* d_ws, size_t ws_size,
                              hipStream_t stream)
{
    (void)in_sizes; (void)n_in; (void)out_size; (void)ws_size;
    const float* x = (const float*)d_in[0];            // (4,7,256,256)
    const float* P[31];
    for (int i = 1; i < 31; ++i) P[i] = (const float*)d_in[i];

    float* ws = (float*)d_ws;
    // Workspace layout (floats). Totals ~79.7M floats (~305 MB).
    float* f0   = ws;                 // 4*32*256*256 = 8388608   (reused as u1)
    float* tmp1 = f0   + 8388608;     // 4*32*128*128 = 2097152   (reused as u2)
    float* d1   = tmp1 + 2097152;     // 2097152
    float* tmp2 = d1   + 2097152;     // 4*32*64*64   = 524288    (reused as u2a)
    float* d2   = tmp2 + 524288;      // 524288
    float* t1   = d2   + 524288;      // 4*32*32*32   = 131072
    float* t2   = t1   + 131072;      // 131072
    float* u0   = t2   + 131072;      // 524288
    float* cat2 = u0   + 524288;      // 4*64*64*64   = 1048576
    float* cat1 = cat2 + 1048576;     // 4*64*128*128 = 4194304
    float* u1a  = cat1 + 4194304;     // 2097152
    float* outc = u1a  + 2097152;     // 8388608
    float* offm = outc + 8388608;     // 4*189*256*256 = 49545216
    float* u2a = tmp2;
    float* u2  = tmp1;
    float* u1  = f0;

    auto cgrid = [](int HoWo, int Cout) {
        return dim3((unsigned)(HoWo / 128), (unsigned)((Cout + 15) / 16), 4u);
    };
    auto ldsb = [](int Cin) {
        int nchk = (Cin + 3) >> 2;
        return (size_t)(nchk * 36 * 16 + 36 * RSB) * sizeof(float);
    };

    // Encoder
    conv3x3_wmma<1,false><<<cgrid(65536,32), TPB, ldsb(7),  stream>>>(x,    P[1],  P[2],  f0,   7, 256,256, 32, 65536, 8);
    conv3x3_wmma<2,true ><<<cgrid(16384,32), TPB, ldsb(32), stream>>>(f0,   P[3],  P[4],  tmp1, 32,256,256, 32, 16384, 7);
    conv3x3_wmma<1,false><<<cgrid(16384,32), TPB, ldsb(32), stream>>>(tmp1, P[5],  P[6],  d1,   32,128,128, 32, 16384, 7);
    conv3x3_wmma<2,true ><<<cgrid(4096,32),  TPB, ldsb(32), stream>>>(d1,   P[7],  P[8],  tmp2, 32,128,128, 32, 4096,  6);
    conv3x3_wmma<1,false><<<cgrid(4096,32),  TPB, ldsb(32), stream>>>(tmp2, P[9],  P[10], d2,   32,64,64,   32, 4096,  6);
    // Bottleneck
    conv3x3_wmma<2,true ><<<cgrid(1024,32),  TPB, ldsb(32), stream>>>(d2,   P[11], P[12], t1,   32,64,64,   32, 1024,  5);
    conv3x3_wmma<1,true ><<<cgrid(1024,32),  TPB, ldsb(32), stream>>>(t1,   P[13], P[14], t2,   32,32,32,   32, 1024,  5);
    convT4x4<<<dim3(524288/TPB), TPB, 0, stream>>>(t2, P[15], P[16], u0, 4, 32, 32, 32, 32);
    // Decoder level 2
    concat2<<<dim3((unsigned)((4*64*4096 + TPB-1)/TPB)), TPB, 0, stream>>>(u0, d2, cat2, 32, 32, 4096, 4);
    conv3x3_wmma<1,true ><<<cgrid(4096,32),  TPB, ldsb(64), stream>>>(cat2, P[17], P[18], u2a, 64,64,64, 32, 4096, 6);
    convT4x4<<<dim3(2097152/TPB), TPB, 0, stream>>>(u2a, P[19], P[20], u2, 4, 32, 64, 64, 32);
    // Decoder level 1
    concat2<<<dim3((unsigned)((4*64*16384 + TPB-1)/TPB)), TPB, 0, stream>>>(u2, d1, cat1, 32, 32, 16384, 4);
    conv3x3_wmma<1,true ><<<cgrid(16384,32), TPB, ldsb(64), stream>>>(cat1, P[21], P[22], u1a, 64,128,128, 32, 16384, 7);
    convT4x4<<<dim3(8388608/TPB), TPB, 0, stream>>>(u1a, P[23], P[24], u1, 4, 32, 128, 128, 32);
    // Heads
    conv3x3_wmma<1,false><<<cgrid(65536,32),  TPB, ldsb(32), stream>>>(u1,   P[25], P[26], outc, 32,256,256, 32,  65536, 8);
    conv3x3_wmma<1,false><<<cgrid(65536,189), TPB, ldsb(32), stream>>>(outc, P[27], P[28], offm, 32,256,256, 189, 65536, 8);
    // Deformable sample + deform-conv einsum
    deform_out<<<dim3(4*65536/TPB), TPB, 0, stream>>>(x, offm, P[29], P[30], (float*)d_out, 4, 256, 256);
}